// SelfSupervisedOrderingLoss_68384469287492
// MI455X (gfx1250) — compile-verified
//
#include <hip/hip_runtime.h>
#include <hip/hip_bf16.h>

// Problem constants (match reference)
#define BCLOUD   2
#define MPTS     8192
#define NPTS     (BCLOUD * MPTS)   // 16384
#define KNEAR    8
#define KFAR     16
#define KSMOOTH  8
// T_LOC = 0.1 -> *10 ; T_CON = 0.5 -> *2
#define INV_TLOC 10.0f
#define INV_TCON 2.0f
#define EPSV     1e-8f

typedef __attribute__((ext_vector_type(2))) float v2f;
typedef __attribute__((ext_vector_type(8))) float v8f;

#define ROWSTRIDE 36   // 32 cols + 4 pad floats; 144B keeps rows 16B-aligned

// ---------------------------------------------------------------------------
// Kernel 1: brute-force kNN per cloud.
// V_WMMA_F32_16X16X4_F32 computes a 16x16 tile of (dot_ij - 0.5*||c_j||^2)
// by packing 1.0 into A's K3 slot and -0.5*csq into B's K3 row, so
// d2_ij = ||q_i||^2 - 2*D_ij with no separate norm broadcast.
// One wave owns 32 query rows; each iteration handles 2 candidate tiles
// (32 candidates, 4 WMMAs), spills D to a per-wave LDS region, waits on
// dscnt once, batch-loads the lane's 32-wide row into 8 float4 registers
// (one DS clause), then runs 32 macro-expanded straight-line top-16 updates
// (no rolled loop -> no v_movrels, no phi copies).
// ---------------------------------------------------------------------------

// One top-16 update step; straight-line, statically indexed throughout.
#define KNN_UPDATE(VAL, COFS)                                              \
  {                                                                        \
    const float d2_ = fmaf(-2.0f, (VAL), qsq);                             \
    if (d2_ < wmax) {                                                      \
      const int j_ = jbase + (COFS);                                       \
      _Pragma("unroll")                                                    \
      for (int k_ = 0; k_ < KFAR; ++k_)                                    \
        if (k_ == wpos) { bd[k_] = d2_; bi[k_] = j_; }                     \
      wmax = bd[0]; wpos = 0;                                              \
      _Pragma("unroll")                                                    \
      for (int k_ = 1; k_ < KFAR; ++k_)                                    \
        if (bd[k_] > wmax) { wmax = bd[k_]; wpos = k_; }                   \
    }                                                                      \
  }

__global__ __launch_bounds__(256) void knn_wmma_kernel(
    const float* __restrict__ coords, int* __restrict__ knn_out) {

  const int lane        = threadIdx.x & 31;
  const int waveInBlock = threadIdx.x >> 5;
  const int waveGlobal  = blockIdx.x * (blockDim.x >> 5) + waveInBlock;
  const int rowBase     = waveGlobal * 32;          // 32 query rows per wave
  const int cloud       = rowBase / MPTS;           // 8192 % 32 == 0 -> uniform
  const int cbase       = cloud * MPTS;

  // per-wave LDS region: 32 rows x 32 cols (stride 36), 16B-aligned
  __shared__ __align__(16) float tile[8][32 * ROWSTRIDE];
  float* my = &tile[waveInBlock][0];

  const int  lsub = lane & 15;
  const bool hi   = lane >= 16;

  // A operands: 16x4 f32 layout — lanes 0-15 hold {K0,K1}, lanes 16-31 {K2,K3}
  // K3 of every query row is 1.0 (pairs with B's -0.5*csq row).
  const float* qa = coords + 3 * (rowBase + lsub);        // rows 0..15
  const float* qb = coords + 3 * (rowBase + 16 + lsub);   // rows 16..31
  v2f Alo, Ahi;
  Alo.x = hi ? qa[2] : qa[0];
  Alo.y = hi ? 1.0f  : qa[1];
  Ahi.x = hi ? qb[2] : qb[0];
  Ahi.y = hi ? 1.0f  : qb[1];

  // my own query row's squared norm (row = rowBase + lane)
  const float* qm = coords + 3 * (rowBase + lane);
  const float qx = qm[0], qy = qm[1], qz = qm[2];
  const float qsq = qx * qx + qy * qy + qz * qz;

  // register top-16 (unsorted; track current worst)
  float bd[KFAR];
  int   bi[KFAR];
#pragma unroll
  for (int k = 0; k < KFAR; ++k) { bd[k] = 3.4e38f; bi[k] = 0x7fffffff; }
  float wmax = 3.4e38f;
  int   wpos = 0;

  // software pipeline: preload candidate coords for the first iteration
  const float* p0 = coords + 3 * (cbase + lsub);
  const float* p1 = coords + 3 * (cbase + 16 + lsub);
  float a0 = p0[0], a1 = p0[1], a2 = p0[2];
  float b0 = p1[0], b1 = p1[1], b2 = p1[2];

  const float4* rowp = reinterpret_cast<const float4*>(my + lane * ROWSTRIDE);

  for (int t = 0; t < MPTS; t += 32) {
    // issue next iteration's loads early (wrap keeps addresses in-bounds)
    const int tn = (t + 32 < MPTS) ? (t + 32) : 0;
    const float* n0 = coords + 3 * (cbase + tn + lsub);
    const float* n1 = coords + 3 * (cbase + tn + 16 + lsub);
    const float na0 = n0[0], na1 = n0[1], na2 = n0[2];
    const float nb0 = n1[0], nb1 = n1[1], nb2 = n1[2];

    // B operands: lanes 0-15 {x,y}; lanes 16-31 {z, -0.5*csq}
    v2f B0, B1;
    B0.x = hi ? a2 : a0;
    B0.y = hi ? -0.5f * (a0 * a0 + a1 * a1 + a2 * a2) : a1;
    B1.x = hi ? b2 : b0;
    B1.y = hi ? -0.5f * (b0 * b0 + b1 * b1 + b2 * b2) : b1;

    v8f D00 = {}, D10 = {}, D01 = {}, D11 = {};
    D00 = __builtin_amdgcn_wmma_f32_16x16x4_f32(false, Alo, false, B0,
                                                (short)0, D00, false, false);
    D10 = __builtin_amdgcn_wmma_f32_16x16x4_f32(false, Ahi, false, B0,
                                                (short)0, D10, false, false);
    D01 = __builtin_amdgcn_wmma_f32_16x16x4_f32(false, Alo, false, B1,
                                                (short)0, D01, false, false);
    D11 = __builtin_amdgcn_wmma_f32_16x16x4_f32(false, Ahi, false, B1,
                                                (short)0, D11, false, false);

    // spill D tiles: vgpr r -> row (r [+8 if hi] [+16 for Ahi]), col per tile
    const int roff = hi ? 8 : 0;
#pragma unroll
    for (int r = 0; r < 8; ++r) my[(r + roff) * ROWSTRIDE + lsub]           = D00[r];
#pragma unroll
    for (int r = 0; r < 8; ++r) my[(r + roff + 16) * ROWSTRIDE + lsub]      = D10[r];
#pragma unroll
    for (int r = 0; r < 8; ++r) my[(r + roff) * ROWSTRIDE + 16 + lsub]      = D01[r];
#pragma unroll
    for (int r = 0; r < 8; ++r) my[(r + roff + 16) * ROWSTRIDE + 16 + lsub] = D11[r];

    // cross-lane store->load ordering within the wave
    asm volatile("s_wait_dscnt 0" ::: "memory");

    // batch-load the lane's whole 32-wide row: 8x ds_load_b128, one wait
    const float4 r0 = rowp[0], r1 = rowp[1], r2 = rowp[2], r3 = rowp[3];
    const float4 r4 = rowp[4], r5 = rowp[5], r6 = rowp[6], r7 = rowp[7];

    const int jbase = cbase + t;
    // 32 straight-line, macro-expanded updates on the float4 components
    KNN_UPDATE(r0.x,  0) KNN_UPDATE(r0.y,  1) KNN_UPDATE(r0.z,  2) KNN_UPDATE(r0.w,  3)
    KNN_UPDATE(r1.x,  4) KNN_UPDATE(r1.y,  5) KNN_UPDATE(r1.z,  6) KNN_UPDATE(r1.w,  7)
    KNN_UPDATE(r2.x,  8) KNN_UPDATE(r2.y,  9) KNN_UPDATE(r2.z, 10) KNN_UPDATE(r2.w, 11)
    KNN_UPDATE(r3.x, 12) KNN_UPDATE(r3.y, 13) KNN_UPDATE(r3.z, 14) KNN_UPDATE(r3.w, 15)
    KNN_UPDATE(r4.x, 16) KNN_UPDATE(r4.y, 17) KNN_UPDATE(r4.z, 18) KNN_UPDATE(r4.w, 19)
    KNN_UPDATE(r5.x, 20) KNN_UPDATE(r5.y, 21) KNN_UPDATE(r5.z, 22) KNN_UPDATE(r5.w, 23)
    KNN_UPDATE(r6.x, 24) KNN_UPDATE(r6.y, 25) KNN_UPDATE(r6.z, 26) KNN_UPDATE(r6.w, 27)
    KNN_UPDATE(r7.x, 28) KNN_UPDATE(r7.y, 29) KNN_UPDATE(r7.z, 30) KNN_UPDATE(r7.w, 31)

    // rotate pipeline registers (same-wave DS ops are in-order: no extra wait)
    a0 = na0; a1 = na1; a2 = na2;
    b0 = nb0; b1 = nb1; b2 = nb2;
  }

  // sort the 16 survivors ascending by (d2, idx) — unrolled bubble network
#pragma unroll
  for (int i = 0; i < KFAR - 1; ++i) {
#pragma unroll
    for (int k = 0; k < KFAR - 1; ++k) {
      if (k < KFAR - 1 - i) {
        const bool sw = (bd[k] > bd[k + 1]) ||
                        (bd[k] == bd[k + 1] && bi[k] > bi[k + 1]);
        const float td = bd[k]; const int ti = bi[k];
        if (sw) { bd[k] = bd[k + 1]; bi[k] = bi[k + 1];
                  bd[k + 1] = td;    bi[k + 1] = ti; }
      }
    }
  }

  const int row = rowBase + lane;
#pragma unroll
  for (int k = 0; k < KFAR; ++k) knn_out[row * KFAR + k] = bi[k];
}

// ---------------------------------------------------------------------------
// Kernel 2: per-point loss terms (loc / contrastive / smooth) with a
// deterministic per-block LDS tree reduction into partials[block*8 + q].
//   q0 = sum w*diff^2, q1 = sum w, q2 = sum log(sig_pos+eps),
//   q3 = sum log(1-sig_neg+eps), q4 = sum (s - mean8)^2
// ---------------------------------------------------------------------------
__global__ __launch_bounds__(256) void loss_pt_kernel(
    const float* __restrict__ scores, const float* __restrict__ coords,
    const int* __restrict__ knn, float* __restrict__ partials) {

  const int i = blockIdx.x * blockDim.x + threadIdx.x;  // < NPTS
  const float s  = scores[i];
  const float x0 = coords[3 * i + 0];
  const float y0 = coords[3 * i + 1];
  const float z0 = coords[3 * i + 2];

  float wsd = 0.0f, wsum = 0.0f, lpos = 0.0f, lneg = 0.0f, smsum = 0.0f;

#pragma unroll
  for (int k = 0; k < KFAR; ++k) {
    const int   j    = knn[i * KFAR + k];
    const float sj   = scores[j];
    const float diff = fabsf(s - sj);
    const float sim  = 1.0f - diff;
    const float sg   = 1.0f / (1.0f + expf(-sim * INV_TCON));
    if (k < KNEAR) {
      const float dx = x0 - coords[3 * j + 0];
      const float dy = y0 - coords[3 * j + 1];
      const float dz = z0 - coords[3 * j + 2];
      const float dist = sqrtf(dx * dx + dy * dy + dz * dz);
      const float w = expf(-dist * INV_TLOC);
      wsum += w;
      wsd  += w * diff * diff;
      lpos += logf(sg + EPSV);
      smsum += sj;
    } else {
      lneg += logf(1.0f - sg + EPSV);
    }
  }
  const float dm = s - smsum * (1.0f / (float)KSMOOTH);
  const float smooth = dm * dm;

  __shared__ float red[256];
  float vals[5] = { wsd, wsum, lpos, lneg, smooth };
#pragma unroll
  for (int q = 0; q < 5; ++q) {
    red[threadIdx.x] = vals[q];
    __syncthreads();
    for (int off = 128; off > 0; off >>= 1) {
      if ((int)threadIdx.x < off) red[threadIdx.x] += red[threadIdx.x + off];
      __syncthreads();
    }
    if (threadIdx.x == 0) partials[blockIdx.x * 8 + q] = red[0];
    __syncthreads();
  }
}

// ---------------------------------------------------------------------------
// Kernel 3: distribution loss via rank counting (replaces per-cloud sort):
// rank(s_i) = #{ s_j < s_i } within the cloud; contrib = (s - rank/(M-1))^2.
// ---------------------------------------------------------------------------
__global__ __launch_bounds__(256) void rank_kernel(
    const float* __restrict__ scores, float* __restrict__ partialsC) {

  const int i     = blockIdx.x * blockDim.x + threadIdx.x;  // < NPTS
  const int cbase = (i / MPTS) * MPTS;
  const float s   = scores[i];

  __shared__ float sh[256];
  int cnt = 0;
  for (int t = 0; t < MPTS; t += 256) {
    sh[threadIdx.x] = scores[cbase + t + threadIdx.x];
    __syncthreads();
#pragma unroll 8
    for (int k = 0; k < 256; ++k) cnt += (sh[k] < s) ? 1 : 0;
    __syncthreads();
  }
  const float tgt = (float)cnt * (1.0f / (float)(MPTS - 1));
  const float d   = s - tgt;
  float v = d * d;

  __shared__ float red[256];
  red[threadIdx.x] = v;
  __syncthreads();
  for (int off = 128; off > 0; off >>= 1) {
    if ((int)threadIdx.x < off) red[threadIdx.x] += red[threadIdx.x + off];
    __syncthreads();
  }
  if (threadIdx.x == 0) partialsC[blockIdx.x] = red[0];
}

// ---------------------------------------------------------------------------
// Kernel 4: finalize — deterministic serial combine of 64+64 partials.
// ---------------------------------------------------------------------------
__global__ void finalize_kernel(const float* __restrict__ pB,
                                const float* __restrict__ pC,
                                float* __restrict__ out, int nblocks) {
  if (threadIdx.x == 0 && blockIdx.x == 0) {
    float S0 = 0, S1 = 0, S2 = 0, S3 = 0, S4 = 0, S6 = 0;
    for (int b = 0; b < nblocks; ++b) {
      S0 += pB[b * 8 + 0];
      S1 += pB[b * 8 + 1];
      S2 += pB[b * 8 + 2];
      S3 += pB[b * 8 + 3];
      S4 += pB[b * 8 + 4];
      S6 += pC[b];
    }
    const float nE = (float)NPTS * (float)KNEAR;
    const float l_loc    = S0 / fmaxf(S1, EPSV);
    const float l_con    = (-S2 / nE) + (-S3 / nE);
    const float l_dist   = S6 / (float)NPTS;   // mean over B of mean over M
    const float l_smooth = S4 / (float)NPTS;
    out[0] = 1.0f * l_loc + 0.5f * l_con + 0.3f * l_dist + 0.2f * l_smooth;
  }
}

// ---------------------------------------------------------------------------
extern "C" void kernel_launch(void* const* d_in, const int* in_sizes, int n_in,
                              void* d_out, int out_size, void* d_ws, size_t ws_size,
                              hipStream_t stream) {
  (void)in_sizes; (void)n_in; (void)out_size; (void)ws_size;
  const float* scores = (const float*)d_in[0];
  const float* coords = (const float*)d_in[1];
  // d_in[2] = batch_ids, unused (deterministic layout: point i -> cloud i/M)
  float* out = (float*)d_out;

  char* ws   = (char*)d_ws;
  int*   knn = (int*)ws;                                   // NPTS*16 ints = 1 MB
  float* pB  = (float*)(ws + (size_t)NPTS * KFAR * sizeof(int)); // 64*8 floats
  float* pC  = pB + 64 * 8;                                      // 64 floats

  knn_wmma_kernel<<<64, 256, 0, stream>>>(coords, knn);
  loss_pt_kernel <<<64, 256, 0, stream>>>(scores, coords, knn, pB);
  rank_kernel    <<<64, 256, 0, stream>>>(scores, pC);
  finalize_kernel<<<1, 32, 0, stream>>>(pB, pC, out, 64);
}